// ProteinFeatures_13864154431837
// MI455X (gfx1250) — compile-verified
//
#include <hip/hip_runtime.h>
#include <hip/hip_bf16.h>
#include <math.h>

typedef __attribute__((ext_vector_type(2))) float v2f;
typedef __attribute__((ext_vector_type(8))) float v8f;

#define TOPK   30
#define NRBF   16
#define NEDGE  25
#define FDIM   416   // 16 pos + 25*16 rbf
#define ODIM   128
#define MTILE  32    // edges per block (2 WMMA M-tiles sharing B fragments)

__constant__ int c_eA[NEDGE] = {1,0,2,3,4,1,1,1,1,0,0,0,4,4,3,0,2,3,4,2,3,4,2,3,2};
__constant__ int c_eB[NEDGE] = {1,0,2,3,4,0,2,3,4,2,3,4,2,3,2,1,1,1,1,0,0,0,4,4,3};

// ---------------------------------------------------------------------------
// Kernel 0: build 5-atom table Y (L,5,3) (atom 4 = virtual Cb) + compact Ca.
// ---------------------------------------------------------------------------
__global__ __launch_bounds__(256) void build_Y_kernel(const float* __restrict__ X,
                                                      float* __restrict__ Yl,
                                                      float* __restrict__ Ca, int L) {
    int l = blockIdx.x * 256 + threadIdx.x;
    if (l >= L) return;
    const float* x = X + (long)l * 12;
    float y0x=x[0],  y0y=x[1],  y0z=x[2];
    float y1x=x[3],  y1y=x[4],  y1z=x[5];
    float y2x=x[6],  y2y=x[7],  y2z=x[8];
    float y3x=x[9],  y3y=x[10], y3z=x[11];
    float bx=y1x-y0x, by=y1y-y0y, bz=y1z-y0z;
    float cx=y2x-y1x, cy=y2y-y1y, cz=y2z-y1z;
    float crx = by*cz - bz*cy;
    float cry = bz*cx - bx*cz;
    float crz = bx*cy - by*cx;
    const float Ac=-0.58273431f, Bc=0.56802827f, Cc=-0.54067466f;
    float cbx = Ac*crx + Bc*bx + Cc*cx + y1x;
    float cby = Ac*cry + Bc*by + Cc*cy + y1y;
    float cbz = Ac*crz + Bc*bz + Cc*cz + y1z;
    float* y = Yl + (long)l * 15;
    y[0]=y0x;  y[1]=y0y;  y[2]=y0z;
    y[3]=y1x;  y[4]=y1y;  y[5]=y1z;
    y[6]=y2x;  y[7]=y2y;  y[8]=y2z;
    y[9]=y3x;  y[10]=y3y; y[11]=y3z;
    y[12]=cbx; y[13]=cby; y[14]=cbz;
    float* ca = Ca + (long)l * 3;
    ca[0]=y1x; ca[1]=y1y; ca[2]=y1z;
}

// ---------------------------------------------------------------------------
// Kernel 1: exact top-30 nearest neighbors per row (one block per row).
// Per-thread candidates in registers; 30 rounds of min-extraction using
// wave32 shfl_xor reduction + one 8-entry LDS merge (2 barriers/round).
// ---------------------------------------------------------------------------
__global__ __launch_bounds__(256) void topk_kernel(const float* __restrict__ Ca,
                                                   const float* __restrict__ mask,
                                                   int* __restrict__ eidx,
                                                   float* __restrict__ idx_out, int L) {
    int i = blockIdx.x;
    int tid = threadIdx.x;
    __shared__ float sv[8];
    __shared__ int   si[8];
    __shared__ int   s_chosen;

    float cix = Ca[(long)i*3 + 0];
    float ciy = Ca[(long)i*3 + 1];
    float ciz = Ca[(long)i*3 + 2];
    float mi = mask[i];

    float dl[32];
#pragma unroll
    for (int q = 0; q < 32; q++) {
        int j = tid + 256 * q;
        float d = 3.0e38f;
        if (j < L) {
            float dx = Ca[(long)j*3 + 0] - cix;
            float dy = Ca[(long)j*3 + 1] - ciy;
            float dz = Ca[(long)j*3 + 2] - ciz;
            d = sqrtf(dx*dx + dy*dy + dz*dz + 1e-6f);
            if (mi * mask[j] == 0.0f) d = 1.0e30f;   // masked pairs last
        }
        dl[q] = d;
    }

    int w = tid >> 5;
    int lane = tid & 31;
    unsigned sel = 0u;
    for (int t = 0; t < TOPK; t++) {
        float bv = 3.0e38f;
        int bj = 0x7fffffff;
#pragma unroll
        for (int q = 0; q < 32; q++) {
            float d = ((sel >> q) & 1u) ? 3.0e38f : dl[q];
            if (d < bv) { bv = d; bj = tid + 256 * q; }
        }
        // wave32 reduction (no barriers)
#pragma unroll
        for (int off = 16; off > 0; off >>= 1) {
            float ov = __shfl_xor(bv, off, 32);
            int   oj = __shfl_xor(bj, off, 32);
            if (ov < bv || (ov == bv && oj < bj)) { bv = ov; bj = oj; }
        }
        if (lane == 0) { sv[w] = bv; si[w] = bj; }
        __syncthreads();
        if (tid == 0) {
            float mv = sv[0]; int mj = si[0];
#pragma unroll
            for (int u = 1; u < 8; u++) {
                float ov = sv[u]; int oj = si[u];
                if (ov < mv || (ov == mv && oj < mj)) { mv = ov; mj = oj; }
            }
            s_chosen = mj;
            eidx[(long)i * TOPK + t] = mj;
            idx_out[(long)i * TOPK + t] = (float)mj;  // exact: mj < 2^24
        }
        __syncthreads();
        int jb = s_chosen;
        if ((jb & 255) == tid) sel |= 1u << (jb >> 8);
    }
}

// ---------------------------------------------------------------------------
// Kernel 2: edge features -> fp32 WMMA GEMM (M=32 edges, N=128, K=416) -> LN.
// 256 threads = 8 waves; wave w owns 16-column N-tile; each B fragment pair
// feeds two V_WMMA_F32_16X16X4_F32 (edge rows 0-15 and 16-31).
// ---------------------------------------------------------------------------
__global__ __launch_bounds__(256) void edge_kernel(
    const float* __restrict__ Yl, const int* __restrict__ eidx,
    const int* __restrict__ ridx, const int* __restrict__ cidx,
    const float* __restrict__ pe_w, const float* __restrict__ pe_b,
    const float* __restrict__ edge_w,
    const float* __restrict__ ln_scale, const float* __restrict__ ln_offset,
    float* __restrict__ out, long nedges)
{
    __shared__ float smem[MTILE * FDIM];   // 32x416 A-tile (53.2 KB); C-tile overlaid
    __shared__ float red[MTILE][8];
    __shared__ float s_mu[MTILE], s_rstd[MTILE];

    float (*feat)[FDIM]  = (float (*)[FDIM])smem;
    float (*ctile)[ODIM] = (float (*)[ODIM])smem;   // reuse after GEMM (barrier-separated)

    int tid = threadIdx.x;
    int s = tid & 15;

    // ---- feature build: 16 threads per edge, two edge halves ----
#pragma unroll
    for (int mh = 0; mh < 2; mh++) {
        int m = (tid >> 4) + 16 * mh;
        long gid = (long)blockIdx.x * MTILE + m;
        long g = (gid < nedges) ? gid : 0;
        int i = (int)(g / TOPK);
        int j = eidx[g];

        // positional-encoding channels [0,16)
        {
            int off = ridx[i] - ridx[j];
            int same = (cidx[i] == cidx[j]) ? 1 : 0;
            int tt = off + 32;
            tt = tt < 0 ? 0 : (tt > 64 ? 64 : tt);
            int dpos = same ? tt : 65;
            feat[m][s] = pe_w[dpos * 16 + s] + pe_b[s];
        }
        // RBF channels [16,416): sub-lane s handles edge types s and s+16
        const float inv_sig = 0.8f;                 // 1 / (20/16)
#pragma unroll
        for (int h = 0; h < 2; h++) {
            int e = s + 16 * h;
            if (e < NEDGE) {
                int a = c_eA[e], b = c_eB[e];
                const float* pa = Yl + (long)i * 15 + a * 3;
                const float* pb = Yl + (long)j * 15 + b * 3;
                float dx = pa[0]-pb[0], dy = pa[1]-pb[1], dz = pa[2]-pb[2];
                float Dn = sqrtf(dx*dx + dy*dy + dz*dz + 1e-6f);
#pragma unroll
                for (int r = 0; r < NRBF; r++) {
                    float mu = 2.0f + (20.0f / 15.0f) * (float)r;
                    float z = (Dn - mu) * inv_sig;
                    feat[m][16 + e * 16 + r] = __expf(-z * z);
                }
            }
        }
    }
    __syncthreads();

    // ---- WMMA GEMM: two 16x16 D tiles per wave, K=416 in steps of 4 ----
    int lane = tid & 31;
    int w = tid >> 5;
    int half = lane >> 4;                 // 0: K pair {0,1}; 1: K pair {2,3}
    int col = w * 16 + (lane & 15);       // global output column
    int arow = lane & 15;                 // A row (M) within tile

    v8f acc0 = {0.f,0.f,0.f,0.f,0.f,0.f,0.f,0.f};
    v8f acc1 = {0.f,0.f,0.f,0.f,0.f,0.f,0.f,0.f};
    for (int k0 = 0; k0 < FDIM; k0 += 4) {
        int ka = k0 + 2 * half;
        v2f b;
        b[0] = edge_w[(long)ka * ODIM + col];
        b[1] = edge_w[(long)(ka + 1) * ODIM + col];
        v2f a0 = *reinterpret_cast<const v2f*>(&feat[arow][ka]);        // edges 0-15
        v2f a1 = *reinterpret_cast<const v2f*>(&feat[arow + 16][ka]);   // edges 16-31
        acc0 = __builtin_amdgcn_wmma_f32_16x16x4_f32(
            false, a0, false, b, (short)0, acc0, false, false);
        acc1 = __builtin_amdgcn_wmma_f32_16x16x4_f32(
            false, a1, false, b, (short)0, acc1, false, false);
    }
    __syncthreads();   // all feat reads complete before C-tile overlays the A-tile
#pragma unroll
    for (int v = 0; v < 8; v++) {
        ctile[v + 8 * half][col]      = acc0[v];   // M = v + 8*half
        ctile[16 + v + 8 * half][col] = acc1[v];
    }
    __syncthreads();

    // ---- LayerNorm over 128 features per edge (two-pass) ----
    int r  = tid >> 3;    // row 0..31
    int c8 = tid & 7;     // 8 threads per row
    float part = 0.f;
#pragma unroll
    for (int q = 0; q < 16; q++) part += ctile[r][c8 + 8 * q];
    red[r][c8] = part;
    __syncthreads();
    if (c8 == 0) {
        float tot = 0.f;
#pragma unroll
        for (int u = 0; u < 8; u++) tot += red[r][u];
        s_mu[r] = tot * (1.0f / ODIM);
    }
    __syncthreads();
    float mu = s_mu[r];

    float p2 = 0.f;
#pragma unroll
    for (int q = 0; q < 16; q++) {
        float d = ctile[r][c8 + 8 * q] - mu;
        p2 += d * d;
    }
    red[r][c8] = p2;
    __syncthreads();
    if (c8 == 0) {
        float tot = 0.f;
#pragma unroll
        for (int u = 0; u < 8; u++) tot += red[r][u];
        s_rstd[r] = rsqrtf(tot * (1.0f / ODIM) + 1e-5f);
    }
    __syncthreads();
    float rstd = s_rstd[r];

    long gid = (long)blockIdx.x * MTILE + r;
    if (gid < nedges) {
        float* orow = out + gid * ODIM;
#pragma unroll
        for (int q = 0; q < 16; q++) {
            int cc = c8 + 8 * q;
            orow[cc] = (ctile[r][cc] - mu) * rstd * ln_scale[cc] + ln_offset[cc];
        }
    }
}

// ---------------------------------------------------------------------------
extern "C" void kernel_launch(void* const* d_in, const int* in_sizes, int n_in,
                              void* d_out, int out_size, void* d_ws, size_t ws_size,
                              hipStream_t stream) {
    const float* X         = (const float*)d_in[0];
    const float* mask      = (const float*)d_in[1];
    const int*   ridx      = (const int*)  d_in[2];
    const int*   cidx      = (const int*)  d_in[3];
    const float* pe_w      = (const float*)d_in[4];
    const float* pe_b      = (const float*)d_in[5];
    const float* edge_w    = (const float*)d_in[6];
    const float* ln_scale  = (const float*)d_in[7];
    const float* ln_offset = (const float*)d_in[8];

    int L = in_sizes[0] / 12;

    float* E_out   = (float*)d_out;
    float* idx_out = E_out + (size_t)L * TOPK * ODIM;

    // workspace: Y table (L*15 f32) | E_idx (L*30 i32) | Ca (L*3 f32)
    float* Yl   = (float*)d_ws;
    int*   eidx = (int*)  ((char*)d_ws + (size_t)L * 15 * sizeof(float));
    float* Ca   = (float*)((char*)d_ws + (size_t)L * 15 * sizeof(float)
                                       + (size_t)L * TOPK * sizeof(int));

    build_Y_kernel<<<(L + 255) / 256, 256, 0, stream>>>(X, Yl, Ca, L);
    topk_kernel<<<L, 256, 0, stream>>>(Ca, mask, eidx, idx_out, L);

    long nedges = (long)L * TOPK;
    int nblk = (int)((nedges + MTILE - 1) / MTILE);
    edge_kernel<<<nblk, 256, 0, stream>>>(Yl, eidx, ridx, cidx, pe_w, pe_b,
                                          edge_w, ln_scale, ln_offset,
                                          E_out, nedges);
}